// TokenSelector_8555574854045
// MI455X (gfx1250) — compile-verified
//
#include <hip/hip_runtime.h>

typedef float v2f __attribute__((ext_vector_type(2)));
typedef float v8f __attribute__((ext_vector_type(8)));

#define DIM   768
#define DIM2  1536
#define NTOK  4096
#define NBATCH 16
#define TLEN  77
#define NKEEP 2048
#define MT    64          // tokens per workgroup tile
#define NBLK  128         // output columns per N-block
#define KC    32          // K chunk staged per step
#define A2STR 72          // float2 per A pair-row (2*72=144 ≡ 16 mod 64 banks)
#define W2STR 136         // float2 per W pair-row (2*136=272 ≡ 16 mod 64 banks)

// ---------------- Kernel 0: text mean + text-half projection (+bias) ----------------
__global__ __launch_bounds__(256) void text_proj_kernel(
    const float* __restrict__ text, const float* __restrict__ W,
    const float* __restrict__ bias, float* __restrict__ tproj)
{
    __shared__ float tm[DIM];
    const int b = blockIdx.x, t = threadIdx.x;
    for (int d = t; d < DIM; d += 256) {
        float s = 0.f;
        const float* p = text + (size_t)b * TLEN * DIM + d;
        #pragma unroll 7
        for (int l = 0; l < TLEN; ++l) s += p[(size_t)l * DIM];
        tm[d] = s * (1.0f / (float)TLEN);
    }
    __syncthreads();
    for (int o = t; o < DIM; o += 256) {
        const float* wr = W + (size_t)o * DIM2 + DIM;   // second half of fusion_W row
        float acc = bias[o];
        for (int d = 0; d < DIM; ++d) acc += tm[d] * wr[d];
        tproj[b * DIM + o] = acc;
    }
}

// ---------------- Kernel 1: main GEMM via V_WMMA_F32_16X16X4_F32 ----------------
// fused[b, n, o] = sum_d img[b, d, n] * W[o, d]  + tproj[b, o]
// LDS tiles are stored as K-pairs (float2) so each WMMA fragment is one ds_load_b64.
__global__ __launch_bounds__(256) void fusion_gemm_kernel(
    const float* __restrict__ img, const float* __restrict__ W,
    const float* __restrict__ tproj, float* __restrict__ fused)
{
    extern __shared__ v2f sm2[];
    v2f* As2 = sm2;                          // [DIM/2][A2STR]  As2[kp][m] = {A[m][2kp], A[m][2kp+1]}
    v2f* Ws2 = sm2 + (DIM / 2) * A2STR;      // [KC/2][W2STR]   Ws2[kp][o] = {W[o][2kp], W[o][2kp+1]}

    const int b   = blockIdx.y;
    const int n0  = blockIdx.x * MT;
    const int tid = threadIdx.x;

    // Stage full A tile: img[b][d][n0 .. n0+63] -> pair rows (768 d-rows x 16 float4)
    {
        const float* base = img + ((size_t)b * DIM) * NTOK + n0;
        for (int i = tid; i < DIM * 16; i += 256) {
            const int d = i >> 4, c = i & 15;
            float4 v = *(const float4*)(base + (size_t)d * NTOK + c * 4);
            const int kp = d >> 1, sub = d & 1;
            float* row = (float*)(As2 + (size_t)kp * A2STR);
            row[(c * 4 + 0) * 2 + sub] = v.x;
            row[(c * 4 + 1) * 2 + sub] = v.y;
            row[(c * 4 + 2) * 2 + sub] = v.z;
            row[(c * 4 + 3) * 2 + sub] = v.w;
        }
    }

    const int wid   = tid >> 5;
    const int wm    = wid & 3;                 // token-row group: rows [wm*16, wm*16+16)
    const int wn    = wid >> 2;                // column half within N-block (0/1 -> 64 cols)
    const int lane  = tid & 31;
    const int lhalf = lane >> 4;               // lane half selects K sub-pair per ISA layout
    const int l16   = lane & 15;
    const int m     = wm * 16 + l16;           // this lane's A row

    for (int nb = 0; nb < DIM / NBLK; ++nb) {
        v8f acc[4];
        #pragma unroll
        for (int nt = 0; nt < 4; ++nt) acc[nt] = (v8f){0.f,0.f,0.f,0.f,0.f,0.f,0.f,0.f};
        const int o0 = nb * NBLK;

        for (int kc = 0; kc < DIM; kc += KC) {
            __syncthreads();   // prior compute done (also fences initial A staging)
            // Stage W chunk as K-pairs: 128 rows x 8 float4 -> 2 b64 stores each
            for (int i = tid; i < NBLK * 8; i += 256) {
                const int oo = i >> 3, kq = i & 7;
                float4 v = *(const float4*)(W + (size_t)(o0 + oo) * DIM2 + kc + kq * 4);
                Ws2[(kq * 2 + 0) * W2STR + oo] = (v2f){v.x, v.y};
                Ws2[(kq * 2 + 1) * W2STR + oo] = (v2f){v.z, v.w};
            }
            __syncthreads();

            #pragma unroll
            for (int k = 0; k < KC; k += 4) {
                // A fragment 16x4: lane half picks K pair {k,k+1} / {k+2,k+3}; one b64 load
                v2f a = As2[(size_t)(((kc + k) >> 1) + lhalf) * A2STR + m];
                #pragma unroll
                for (int nt = 0; nt < 4; ++nt) {
                    const int oc = wn * 64 + nt * 16 + l16;
                    v2f bb = Ws2[(size_t)((k >> 1) + lhalf) * W2STR + oc];
                    acc[nt] = __builtin_amdgcn_wmma_f32_16x16x4_f32(
                        false, a, false, bb, (short)0, acc[nt], false, false);
                }
            }
        }

        // Epilogue: add text projection (+bias), write fused tile.
        // C layout: VGPR j holds row (j + lhalf*8), column = l16 within the 16-tile.
        #pragma unroll
        for (int nt = 0; nt < 4; ++nt) {
            const int oc = o0 + wn * 64 + nt * 16 + l16;
            const float tb = tproj[b * DIM + oc];
            float* outp = fused + ((size_t)b * NTOK + n0 + wm * 16 + lhalf * 8) * DIM + oc;
            #pragma unroll
            for (int j = 0; j < 8; ++j)
                outp[(size_t)j * DIM] = acc[nt][j] + tb;
        }
    }
}

// ---------------- Kernel 2: LayerNorm + score dot (softmax skipped: monotonic) ------
__global__ __launch_bounds__(256) void ln_score_kernel(
    const float* __restrict__ fused, const float* __restrict__ g,
    const float* __restrict__ bta,   const float* __restrict__ sW,
    const float* __restrict__ sb,    float* __restrict__ scores)
{
    const int token = blockIdx.x * 8 + (threadIdx.x >> 5);
    const int lane  = threadIdx.x & 31;
    const float* row = fused + (size_t)token * DIM;
    float s = 0.f, ss = 0.f;
    for (int i = lane; i < DIM; i += 32) { float v = row[i]; s += v; ss += v * v; }
    #pragma unroll
    for (int off = 16; off; off >>= 1) { s += __shfl_xor(s, off); ss += __shfl_xor(ss, off); }
    const float mu  = s * (1.0f / DIM);
    const float var = ss * (1.0f / DIM) - mu * mu;
    const float inv = rsqrtf(var + 1e-5f);
    float dot = 0.f;
    for (int i = lane; i < DIM; i += 32)
        dot += ((row[i] - mu) * inv * g[i] + bta[i]) * sW[i];
    #pragma unroll
    for (int off = 16; off; off >>= 1) dot += __shfl_xor(dot, off);
    if (lane == 0) scores[token] = dot + sb[0];
}

// ---------------- Kernel 3: exact top-k by rank + index-ordered compaction ----------
__global__ __launch_bounds__(1024) void topk_kernel(
    const float* __restrict__ scores, int* __restrict__ idx)
{
    __shared__ float sc[NTOK];
    __shared__ int   kp[NTOK];
    const int b = blockIdx.x, t = threadIdx.x;
    const float* s = scores + (size_t)b * NTOK;
    for (int i = t; i < NTOK; i += 1024) sc[i] = s[i];
    __syncthreads();
    for (int i = t; i < NTOK; i += 1024) {
        const float v = sc[i];
        int rank = 0;
        for (int j = 0; j < NTOK; ++j) {
            const float u = sc[j];
            rank += (u > v) || (u == v && j < i);   // jax stable descending order
        }
        kp[i] = (rank < NKEEP) ? 1 : 0;
    }
    __syncthreads();
    for (int i = t; i < NTOK; i += 1024) {
        if (kp[i]) {
            int pos = 0;
            for (int j = 0; j < i; ++j) pos += kp[j];
            idx[b * NKEEP + pos] = i;               // pos monotone in i -> sorted
        }
    }
}

// ---------------- Kernel 4: gather selected rows ----------------
__global__ __launch_bounds__(256) void gather_kernel(
    const float* __restrict__ fused, const int* __restrict__ idx,
    float* __restrict__ out)
{
    const int p    = blockIdx.x * 8 + (threadIdx.x >> 5);   // selected row id
    const int lane = threadIdx.x & 31;
    const int b    = p >> 11;          // / NKEEP
    const int pp   = p & (NKEEP - 1);
    const int src  = idx[b * NKEEP + pp];
    const float4* sp = (const float4*)(fused + ((size_t)b * NTOK + src) * DIM);
    float4* dp       = (float4*)(out + (size_t)p * DIM);
    for (int i = lane; i < DIM / 4; i += 32) dp[i] = sp[i];
}

extern "C" void kernel_launch(void* const* d_in, const int* in_sizes, int n_in,
                              void* d_out, int out_size, void* d_ws, size_t ws_size,
                              hipStream_t stream) {
    const float* img  = (const float*)d_in[0];
    const float* text = (const float*)d_in[1];
    const float* fW   = (const float*)d_in[2];
    const float* fb   = (const float*)d_in[3];
    const float* lg   = (const float*)d_in[4];
    const float* lb   = (const float*)d_in[5];
    const float* sW   = (const float*)d_in[6];
    const float* sb   = (const float*)d_in[7];
    float* out = (float*)d_out;

    float* tproj  = (float*)d_ws;                                  // 16*768
    float* fused  = tproj + NBATCH * DIM;                          // 16*4096*768
    float* scores = fused + (size_t)NBATCH * NTOK * DIM;           // 16*4096
    int*   idx    = (int*)(scores + NBATCH * NTOK);                // 16*2048

    const size_t shmem = ((size_t)(DIM / 2) * A2STR + (KC / 2) * W2STR) * sizeof(v2f); // ~233 KB
    (void)hipFuncSetAttribute((const void*)fusion_gemm_kernel,
                              hipFuncAttributeMaxDynamicSharedMemorySize, (int)shmem);

    text_proj_kernel<<<NBATCH, 256, 0, stream>>>(text, fW, fb, tproj);
    dim3 g1(NTOK / MT, NBATCH);
    fusion_gemm_kernel<<<g1, 256, shmem, stream>>>(img, fW, tproj, fused);
    ln_score_kernel<<<(NBATCH * NTOK) / 8, 256, 0, stream>>>(fused, lg, lb, sW, sb, scores);
    topk_kernel<<<NBATCH, 1024, 0, stream>>>(scores, idx);
    gather_kernel<<<(NBATCH * NKEEP) / 8, 256, 0, stream>>>(fused, idx, out);
}